// GCN_21071109554677
// MI455X (gfx1250) — compile-verified
//
#include <hip/hip_runtime.h>
#include <hip/hip_bf16.h>

typedef __attribute__((ext_vector_type(2))) float v2f;
typedef __attribute__((ext_vector_type(8))) float v8f;

#define DFEAT 128
#define LDA   132   // padded LDS row (bank-conflict-free: stride%64 == 4)

// ---------------- zero fill ----------------
__global__ void gcn_zero(float* __restrict__ p, int n) {
    int i = blockIdx.x * blockDim.x + threadIdx.x;
    if (i < n) p[i] = 0.0f;
}

// ---------------- degree counting ----------------
__global__ void gcn_deg(const int* __restrict__ ei,
                        float* __restrict__ outdeg, float* __restrict__ indeg,
                        int n_edges) {
    int e = blockIdx.x * blockDim.x + threadIdx.x;
    if (e >= n_edges) return;
    int src = ei[2 * e];
    int dst = ei[2 * e + 1];
    atomicAdd(&outdeg[src], 1.0f);
    atomicAdd(&indeg[dst], 1.0f);
}

// ---------------- deg -> max(1,deg)^(-1/2), in place ----------------
__global__ void gcn_norm(float* __restrict__ deg, int n) {
    int i = blockIdx.x * blockDim.x + threadIdx.x;
    if (i < n) {
        float d = deg[i];
        d = d < 1.0f ? 1.0f : d;
        deg[i] = 1.0f / sqrtf(d);
    }
}

// ---------------- edge scatter: agg[dst] += x[src] * onorm[src] ----------------
// one wave (32 lanes) per edge; each lane moves a float4 (32*4 = 128 feats)
__global__ void gcn_scatter(const float* __restrict__ x,
                            const int* __restrict__ ei,
                            const float* __restrict__ onorm,
                            float* __restrict__ agg,
                            int n_edges) {
    int t    = blockIdx.x * blockDim.x + threadIdx.x;
    int e    = t >> 5;
    int lane = t & 31;
    if (e >= n_edges) return;
    int src = ei[2 * e];
    int dst = ei[2 * e + 1];
    float s = onorm[src];
    const float4* xr = (const float4*)(x + (size_t)src * DFEAT);
    float4 v = xr[lane];
    float* ar = agg + (size_t)dst * DFEAT + lane * 4;
    atomicAdd(ar + 0, v.x * s);
    atomicAdd(ar + 1, v.y * s);
    atomicAdd(ar + 2, v.z * s);
    atomicAdd(ar + 3, v.w * s);
}

// ---------------- fused GEMM + in_norm scale + ReLU (in place on agg) --------
// block = 256 threads = 8 waves; each wave owns one 16-col N-tile.
// Wave computes C[16x16] = A[16x128] * W[128x16] via 32x V_WMMA_F32_16X16X4_F32.
// A block staged in LDS (read fully before the in-place row writes -> safe).
__global__ void gcn_gemm(float* __restrict__ agg,          // in: agg, out: result
                         const float* __restrict__ W,      // 128x128 row-major
                         const float* __restrict__ innorm,
                         int n_nodes) {
    __shared__ float sA[16 * LDA];   // 8448 B

    const int tid  = threadIdx.x;
    const int wave = tid >> 5;       // 0..7  -> N-tile
    const int lane = tid & 31;
    const size_t m0 = (size_t)blockIdx.x * 16;

    // cooperative load of the 16x128 A block into LDS (zero-pad tail rows)
    for (int i = tid; i < 16 * DFEAT; i += 256) {
        int r = i >> 7;          // 0..15
        int c = i & 127;         // 0..127
        size_t row = m0 + r;
        sA[r * LDA + c] = (row < (size_t)n_nodes) ? agg[row * DFEAT + c] : 0.0f;
    }
    __syncthreads();

    const int M    = lane & 15;      // A row / C col-index / B col-index
    const int half = lane >> 4;      // 0 or 1
    const int n0   = wave * 16;

    v8f acc = {};
    #pragma unroll
    for (int k0 = 0; k0 < DFEAT; k0 += 4) {
        const int ka = k0 + 2 * half;
        // A 16x4: lane -> (M, ka..ka+1), contiguous in LDS
        v2f a;
        const float2 af = *(const float2*)(&sA[M * LDA + ka]);
        a.x = af.x; a.y = af.y;
        // B 4x16: lane -> (ka..ka+1, n0+M), stride-128 rows in W (L2-hot)
        v2f b;
        b.x = W[(size_t)ka * 128 + n0 + M];
        b.y = W[(size_t)(ka + 1) * 128 + n0 + M];
        acc = __builtin_amdgcn_wmma_f32_16x16x4_f32(
            false, a, false, b, (short)0, acc, false, false);
    }

    // epilogue: row = i + 8*half, col = n0 + M ; scale by in_norm, ReLU
    #pragma unroll
    for (int i = 0; i < 8; ++i) {
        size_t row = m0 + i + 8 * half;
        if (row < (size_t)n_nodes) {
            float v = acc[i] * innorm[row];
            agg[row * DFEAT + n0 + M] = v > 0.0f ? v : 0.0f;
        }
    }
}

extern "C" void kernel_launch(void* const* d_in, const int* in_sizes, int n_in,
                              void* d_out, int out_size, void* d_ws, size_t ws_size,
                              hipStream_t stream) {
    const float* x  = (const float*)d_in[0];   // [n_nodes, 128] f32
    const float* W  = (const float*)d_in[1];   // [128, 128] f32
    const int*   ei = (const int*)d_in[2];     // [n_edges, 2] i32
    float* out = (float*)d_out;                // [n_nodes, 128] f32 (doubles as agg)

    const int n_nodes = in_sizes[0] / DFEAT;
    const int n_edges = in_sizes[2] / 2;

    float* outdeg = (float*)d_ws;              // n_nodes floats -> becomes out_norm
    float* indeg  = outdeg + n_nodes;          // n_nodes floats -> becomes in_norm

    const int B = 256;

    // 1) zero agg (= d_out) and both degree arrays
    {
        int n = n_nodes * DFEAT;
        gcn_zero<<<(n + B - 1) / B, B, 0, stream>>>(out, n);
        int m = 2 * n_nodes;
        gcn_zero<<<(m + B - 1) / B, B, 0, stream>>>(outdeg, m);
    }
    // 2) degree counts
    gcn_deg<<<(n_edges + B - 1) / B, B, 0, stream>>>(ei, outdeg, indeg, n_edges);
    // 3) degrees -> norms (both arrays, contiguous, in place)
    {
        int m = 2 * n_nodes;
        gcn_norm<<<(m + B - 1) / B, B, 0, stream>>>(outdeg, m);
    }
    // 4) edge scatter (one wave per edge)
    {
        long long threads = (long long)n_edges * 32;
        int blocks = (int)((threads + B - 1) / B);
        gcn_scatter<<<blocks, B, 0, stream>>>(x, ei, outdeg, out, n_edges);
    }
    // 5) fused WMMA GEMM + in_norm + ReLU, in place on d_out
    {
        int blocks = (n_nodes + 15) / 16;
        gcn_gemm<<<blocks, B, 0, stream>>>(out, W, indeg, n_nodes);
    }
}